// MambaEEGModel_1932735283985
// MI455X (gfx1250) — compile-verified
//
#include <hip/hip_runtime.h>
#include <hip/hip_bf16.h>
#include <math.h>

#define BATCH 64
#define NCH 129
#define NT 200
#define D_MODEL 256
#define D_STATE 16
#define HEADDIM 32
#define NLAYERS 4
#define D_INNER 512
#define NHEADS 16
#define CONV_DIM 544
#define D_IN_PROJ 1072
#define ROWS (BATCH * NT) /* 12800 */

typedef float v2f __attribute__((ext_vector_type(2)));
typedef float v8f __attribute__((ext_vector_type(8)));

__device__ __forceinline__ float silu_f(float x) { return x / (1.f + __expf(-x)); }

static inline int cdiv(long long a, long long b) { return (int)((a + b - 1) / b); }

// ---------------- encoder: depthwise conv over time (k=3, pad 1) ----------------
__global__ void k_sconv(const float* __restrict__ x, const float* __restrict__ w,
                        const float* __restrict__ b, float* __restrict__ xs) {
  int i = blockIdx.x * blockDim.x + threadIdx.x;
  if (i >= BATCH * NCH * NT) return;
  int t = i % NT;
  int c = (i / NT) % NCH;
  const float* xp = x + (i - t);
  float acc = b[c];
#pragma unroll
  for (int k = 0; k < 3; k++) {
    int tt = t - 1 + k;
    if (tt >= 0 && tt < NT) acc += xp[tt] * w[c * 3 + k];
  }
  xs[i] = acc;
}

__global__ void k_embmean(const float* __restrict__ emb, float* __restrict__ em) {
  int d = threadIdx.x;  // 256
  float s = 0.f;
  for (int c = 0; c < NCH; c++) s += emb[c * D_MODEL + d];
  em[d] = s * (1.f / NCH);
}

// h[b,t,d] = sum_c xs[b,c,t] * mixer_w[d,c] + mixer_b[d] + embmean[d]
__global__ void k_mixer(const float* __restrict__ xs, const float* __restrict__ w,
                        const float* __restrict__ bias, const float* __restrict__ em,
                        float* __restrict__ h) {
  int i = blockIdx.x * blockDim.x + threadIdx.x;
  if (i >= ROWS * D_MODEL) return;
  int d = i % D_MODEL;
  int row = i / D_MODEL;
  int bb = row / NT, t = row % NT;
  const float* xp = xs + (size_t)(bb * NCH) * NT + t;
  const float* wp = w + (size_t)d * NCH;
  float acc = bias[d] + em[d];
  for (int c = 0; c < NCH; c++) acc += xp[c * NT] * wp[c];
  h[i] = acc;
}

// ---------------- layernorm over 256, optional residual ----------------
__global__ void k_ln256(const float* __restrict__ src, const float* __restrict__ res,
                        const float* __restrict__ w, const float* __restrict__ bias,
                        float* __restrict__ dst) {
  __shared__ float red[256];
  int row = blockIdx.x;
  int d = threadIdx.x;
  float v = src[(size_t)row * D_MODEL + d];
  if (res) v += res[(size_t)row * D_MODEL + d];
  red[d] = v;
  __syncthreads();
  for (int s = 128; s > 0; s >>= 1) {
    if (d < s) red[d] += red[d + s];
    __syncthreads();
  }
  float mean = red[0] * (1.f / D_MODEL);
  __syncthreads();
  float c = v - mean;
  red[d] = c * c;
  __syncthreads();
  for (int s = 128; s > 0; s >>= 1) {
    if (d < s) red[d] += red[d + s];
    __syncthreads();
  }
  float var = red[0] * (1.f / D_MODEL);
  dst[(size_t)row * D_MODEL + d] = c * rsqrtf(var + 1e-5f) * w[d] + bias[d];
}

// ---------------- WMMA f32 GEMM: C[M,N] = A[M,K] @ W[N,K]^T ----------------
// Register-blocked: one wave computes a 64(M) x 16(N) strip = 4 C tiles.
// Software-pipelined: loads for step k+4 are issued before the WMMAs of step k,
// so ~5 loads stay in flight while the matrix pipe drains (partial loadcnt waits).
// A 16x4 layout (ISA 7.12.2): lanes 0-15 hold M=0..15, K={0,1}; lanes 16-31 K={2,3}.
// B 4x16 mirrors with N across lanes. C/D: VGPR v -> M=v (+8 for lanes>=16), N=lane&15.
#define MBLK 4
#define WMMA4(a, b, c) __builtin_amdgcn_wmma_f32_16x16x4_f32(false, a, false, b, (short)0, c, false, false)
__global__ void k_gemm_wmma(const float* __restrict__ A, const float* __restrict__ W,
                            float* __restrict__ C, int M, int N, int K) {
  int wave = (int)((blockIdx.x * blockDim.x + threadIdx.x) >> 5);
  int lane = threadIdx.x & 31;
  int tN = N >> 4;
  int waves = (M >> 6) * tN;  // M is a multiple of 64
  if (wave >= waves) return;
  int tm = (wave / tN) << 6;
  int tn = (wave % tN) << 4;
  int half = lane >> 4, l = lane & 15;
  int col = tn + l;
  const float* ap = A + (size_t)(tm + l) * K + half * 2;
  const float* bp = W + (size_t)(tn + l) * K + half * 2;
  v8f acc0 = {}, acc1 = {}, acc2 = {}, acc3 = {};
#if defined(__HIP_DEVICE_COMPILE__) && __has_builtin(__builtin_amdgcn_wmma_f32_16x16x4_f32)
  const size_t astep = (size_t)16 * K;
  // prologue: fragments for k = 0
  v2f bv = *(const v2f*)(bp);
  v2f a0 = *(const v2f*)(ap);
  v2f a1 = *(const v2f*)(ap + astep);
  v2f a2 = *(const v2f*)(ap + 2 * astep);
  v2f a3 = *(const v2f*)(ap + 3 * astep);
  for (int k = 4; k < K; k += 4) {
    // issue next-step loads first (distinct live ranges -> stay in flight)
    v2f bn = *(const v2f*)(bp + k);
    v2f n0 = *(const v2f*)(ap + k);
    v2f n1 = *(const v2f*)(ap + astep + k);
    v2f n2 = *(const v2f*)(ap + 2 * astep + k);
    v2f n3 = *(const v2f*)(ap + 3 * astep + k);
    // consume previous step's fragments
    acc0 = WMMA4(a0, bv, acc0);
    acc1 = WMMA4(a1, bv, acc1);
    acc2 = WMMA4(a2, bv, acc2);
    acc3 = WMMA4(a3, bv, acc3);
    bv = bn; a0 = n0; a1 = n1; a2 = n2; a3 = n3;
  }
  // epilogue
  acc0 = WMMA4(a0, bv, acc0);
  acc1 = WMMA4(a1, bv, acc1);
  acc2 = WMMA4(a2, bv, acc2);
  acc3 = WMMA4(a3, bv, acc3);
#else
  // numerically-identical scalar fallback (host pass / missing builtin)
  v8f* accs[MBLK] = {&acc0, &acc1, &acc2, &acc3};
  for (int j = 0; j < MBLK; j++) {
    int row0 = tm + j * 16 + half * 8;
    for (int v = 0; v < 8; v++) {
      float a = 0.f;
      const float* ar = A + (size_t)(row0 + v) * K;
      const float* wr = W + (size_t)col * K;
      for (int k = 0; k < K; k++) a += ar[k] * wr[k];
      (*accs[j])[v] = a;
    }
  }
#endif
  {
    int row0 = tm + half * 8;
#pragma unroll
    for (int v = 0; v < 8; v++) C[(size_t)(row0 + v) * N + col] = acc0[v];
    row0 += 16;
#pragma unroll
    for (int v = 0; v < 8; v++) C[(size_t)(row0 + v) * N + col] = acc1[v];
    row0 += 16;
#pragma unroll
    for (int v = 0; v < 8; v++) C[(size_t)(row0 + v) * N + col] = acc2[v];
    row0 += 16;
#pragma unroll
    for (int v = 0; v < 8; v++) C[(size_t)(row0 + v) * N + col] = acc3[v];
  }
}

// ---------------- causal depthwise conv (k=4) + SiLU on xBC slice ----------------
__global__ void k_convsilu(const float* __restrict__ zx, const float* __restrict__ cw,
                           const float* __restrict__ cb, float* __restrict__ xbc) {
  int i = blockIdx.x * blockDim.x + threadIdx.x;
  if (i >= ROWS * CONV_DIM) return;
  int c = i % CONV_DIM;
  int row = i / CONV_DIM;
  int t = row % NT;
  float acc = cb[c];
#pragma unroll
  for (int k = 0; k < 4; k++) {
    int tt = t - 3 + k;
    if (tt >= 0) acc += zx[(size_t)(row - 3 + k) * D_IN_PROJ + D_INNER + c] * cw[c * 4 + k];
  }
  xbc[i] = silu_f(acc);
}

// ---------------- dt = softplus(raw + dt_bias) ----------------
__global__ void k_dt(const float* __restrict__ zx, const float* __restrict__ dtb,
                     float* __restrict__ dt) {
  int i = blockIdx.x * blockDim.x + threadIdx.x;
  if (i >= ROWS * NHEADS) return;
  int hh = i % NHEADS;
  int row = i / NHEADS;
  float v = zx[(size_t)row * D_IN_PROJ + D_INNER + CONV_DIM + hh] + dtb[hh];
  dt[i] = (v > 20.f) ? v : log1pf(__expf(v));
}

// ---------------- sequential selective-state scan: one wave per (b, head) ----------------
__global__ void k_scan(const float* __restrict__ xbc, const float* __restrict__ dt,
                       const float* __restrict__ Alog, const float* __restrict__ Dsk,
                       float* __restrict__ y) {
  int bh = blockIdx.x;
  int p = threadIdx.x;  // 0..31 (headdim lane)
  int bb = bh >> 4, hh = bh & 15;
  float A = -__expf(Alog[hh]);
  float Dh = Dsk[hh];
  float s[D_STATE];
#pragma unroll
  for (int n = 0; n < D_STATE; n++) s[n] = 0.f;
  const float* xb = xbc + (size_t)(bb * NT) * CONV_DIM;
  const float* dtp = dt + (size_t)(bb * NT) * NHEADS + hh;
  float* yp = y + (size_t)(bb * NT) * D_INNER + hh * HEADDIM + p;
  for (int t = 0; t < NT; t++) {
    const float* xr = xb + (size_t)t * CONV_DIM;
    float dtv = dtp[(size_t)t * NHEADS];
    float dA = __expf(dtv * A);
    float xv = xr[hh * HEADDIM + p];
    float dtx = dtv * xv;
    float4 B0 = *(const float4*)(xr + D_INNER + 0);
    float4 B1 = *(const float4*)(xr + D_INNER + 4);
    float4 B2 = *(const float4*)(xr + D_INNER + 8);
    float4 B3 = *(const float4*)(xr + D_INNER + 12);
    float4 C0 = *(const float4*)(xr + D_INNER + 16);
    float4 C1 = *(const float4*)(xr + D_INNER + 20);
    float4 C2 = *(const float4*)(xr + D_INNER + 24);
    float4 C3 = *(const float4*)(xr + D_INNER + 28);
    float Bv[16] = {B0.x, B0.y, B0.z, B0.w, B1.x, B1.y, B1.z, B1.w,
                    B2.x, B2.y, B2.z, B2.w, B3.x, B3.y, B3.z, B3.w};
    float Cv[16] = {C0.x, C0.y, C0.z, C0.w, C1.x, C1.y, C1.z, C1.w,
                    C2.x, C2.y, C2.z, C2.w, C3.x, C3.y, C3.z, C3.w};
    float acc = 0.f;
#pragma unroll
    for (int n = 0; n < D_STATE; n++) {
      s[n] = dA * s[n] + dtx * Bv[n];
      acc += s[n] * Cv[n];
    }
    yp[(size_t)t * D_INNER] = acc + xv * Dh;
  }
}

// ---------------- gate with silu(z), then RMSNorm over 512 ----------------
__global__ void k_gatenorm(const float* __restrict__ zx, const float* __restrict__ nw,
                           float* __restrict__ y) {
  __shared__ float red[256];
  int row = blockIdx.x;
  int tid = threadIdx.x;
  const float* zr = zx + (size_t)row * D_IN_PROJ;  // z = first 512 cols
  float* yr = y + (size_t)row * D_INNER;
  float g0 = yr[tid] * silu_f(zr[tid]);
  float g1 = yr[tid + 256] * silu_f(zr[tid + 256]);
  red[tid] = g0 * g0 + g1 * g1;
  __syncthreads();
  for (int s = 128; s > 0; s >>= 1) {
    if (tid < s) red[tid] += red[tid + s];
    __syncthreads();
  }
  float scale = rsqrtf(red[0] * (1.f / D_INNER) + 1e-5f);
  yr[tid] = g0 * scale * nw[tid];
  yr[tid + 256] = g1 * scale * nw[tid + 256];
}

// ---------------- mean-pool over time + 2-layer head ----------------
__global__ void k_head(const float* __restrict__ h, const float* __restrict__ w1,
                       const float* __restrict__ b1, const float* __restrict__ w2,
                       const float* __restrict__ b2, float* __restrict__ out) {
  __shared__ float pooled[D_MODEL];
  __shared__ float hid[128];
  int bb = blockIdx.x;
  int tid = threadIdx.x;  // 256
  float s = 0.f;
  const float* hp = h + (size_t)(bb * NT) * D_MODEL + tid;
  for (int t = 0; t < NT; t++) s += hp[(size_t)t * D_MODEL];
  pooled[tid] = s * (1.f / NT);
  __syncthreads();
  if (tid < 128) {
    float a = b1[tid];
    const float* wp = w1 + (size_t)tid * D_MODEL;
    for (int d = 0; d < D_MODEL; d++) a += pooled[d] * wp[d];
    hid[tid] = a > 0.f ? a : 0.f;
  }
  __syncthreads();
  if (tid == 0) {
    float a = b2[0];
    for (int j = 0; j < 128; j++) a += hid[j] * w2[j];
    out[bb] = a;
  }
}

extern "C" void kernel_launch(void* const* d_in, const int* in_sizes, int n_in,
                              void* d_out, int out_size, void* d_ws, size_t ws_size,
                              hipStream_t stream) {
  (void)in_sizes; (void)n_in; (void)out_size; (void)ws_size;
  const float* x = (const float*)d_in[0];
  const float* emb = (const float*)d_in[1];
  const float* sconv_w = (const float*)d_in[2];
  const float* sconv_b = (const float*)d_in[3];
  const float* mixer_w = (const float*)d_in[4];
  const float* mixer_b = (const float*)d_in[5];
  const float* enc_ln_w = (const float*)d_in[6];
  const float* enc_ln_b = (const float*)d_in[7];
  const float* in_proj_w = (const float*)d_in[8];
  const float* conv_w = (const float*)d_in[9];
  const float* conv_b = (const float*)d_in[10];
  const float* dt_bias = (const float*)d_in[11];
  const float* A_log = (const float*)d_in[12];
  const float* D_skip = (const float*)d_in[13];
  const float* norm_w = (const float*)d_in[14];
  const float* out_proj_w = (const float*)d_in[15];
  const float* ln_w = (const float*)d_in[16];
  const float* ln_b = (const float*)d_in[17];
  const float* head1_w = (const float*)d_in[18];
  const float* head1_b = (const float*)d_in[19];
  const float* head2_w = (const float*)d_in[20];
  const float* head2_b = (const float*)d_in[21];

  float* ws = (float*)d_ws;
  size_t off = 0;
  float* f_h = ws + off;   off += (size_t)ROWS * D_MODEL;    // 3.28M floats
  float* f_zx = ws + off;  off += (size_t)ROWS * D_IN_PROJ;  // 13.7M
  float* f_xbc = ws + off; off += (size_t)ROWS * CONV_DIM;   // 6.96M
  float* f_dt = ws + off;  off += (size_t)ROWS * NHEADS;     // 0.20M
  float* f_y = ws + off;   off += (size_t)ROWS * D_INNER;    // 6.55M
  float* f_tmp = ws + off; off += (size_t)ROWS * D_MODEL;    // 3.28M
  float* f_em = ws + off;  off += 256;
  float* f_xs = f_zx;  // encoder scratch aliases zxbcdt (disjoint in time)

  // ---- encoder ----
  k_sconv<<<cdiv((long long)BATCH * NCH * NT, 256), 256, 0, stream>>>(x, sconv_w, sconv_b, f_xs);
  k_embmean<<<1, 256, 0, stream>>>(emb, f_em);
  k_mixer<<<cdiv((long long)ROWS * D_MODEL, 256), 256, 0, stream>>>(f_xs, mixer_w, mixer_b,
                                                                    f_em, f_tmp);
  k_ln256<<<ROWS, 256, 0, stream>>>(f_tmp, nullptr, enc_ln_w, enc_ln_b, f_h);

  // ---- Mamba2 layers ----
  for (int i = 0; i < NLAYERS; i++) {
    long long waves = (long long)(ROWS / 64) * (D_IN_PROJ / 16);
    k_gemm_wmma<<<cdiv(waves * 32, 256), 256, 0, stream>>>(
        f_h, in_proj_w + (size_t)i * D_IN_PROJ * D_MODEL, f_zx, ROWS, D_IN_PROJ, D_MODEL);
    k_convsilu<<<cdiv((long long)ROWS * CONV_DIM, 256), 256, 0, stream>>>(
        f_zx, conv_w + (size_t)i * CONV_DIM * 4, conv_b + (size_t)i * CONV_DIM, f_xbc);
    k_dt<<<cdiv((long long)ROWS * NHEADS, 256), 256, 0, stream>>>(
        f_zx, dt_bias + (size_t)i * NHEADS, f_dt);
    k_scan<<<BATCH * NHEADS, 32, 0, stream>>>(f_xbc, f_dt, A_log + (size_t)i * NHEADS,
                                              D_skip + (size_t)i * NHEADS, f_y);
    k_gatenorm<<<ROWS, 256, 0, stream>>>(f_zx, norm_w + (size_t)i * D_INNER, f_y);
    waves = (long long)(ROWS / 64) * (D_MODEL / 16);
    k_gemm_wmma<<<cdiv(waves * 32, 256), 256, 0, stream>>>(
        f_y, out_proj_w + (size_t)i * D_MODEL * D_INNER, f_tmp, ROWS, D_MODEL, D_INNER);
    k_ln256<<<ROWS, 256, 0, stream>>>(f_tmp, f_h, ln_w + (size_t)i * D_MODEL,
                                      ln_b + (size_t)i * D_MODEL, f_h);
  }

  // ---- pool + head ----
  k_head<<<BATCH, 256, 0, stream>>>(f_h, head1_w, head1_b, head2_w, head2_b, (float*)d_out);
}